// LayerController_58497454571718
// MI455X (gfx1250) — compile-verified
//
#include <hip/hip_runtime.h>
#include <math.h>

typedef __attribute__((ext_vector_type(2))) float v2f;
typedef __attribute__((ext_vector_type(8))) float v8f;

#define DIM   4096
#define NTOK  8192
#define KC    64          // K-chunk staged in LDS (divides 4096)
#define NCOL  160         // 32 (u) + 64 (h) + 64 (g) output columns in phase 1
#define WSTR  68          // LDS row stride (floats): 68*4B, bank-conflict-free, 16B aligned

#define WMMA_F32(A, B, C) \
  __builtin_amdgcn_wmma_f32_16x16x4_f32(false, (A), false, (B), (short)0, (C), false, false)

// ---------------------------------------------------------------------------
// Kernel A: C1 = W1[:,4096:] @ B_bank  (64x16 each, 4 matrices)
// C1[j][r<16] uses B_pos, [r>=16] uses B_neg; same for Cg1 with G1.
// ---------------------------------------------------------------------------
__global__ __launch_bounds__(256) void cmat_kernel(
    const float* __restrict__ W1, const float* __restrict__ G1,
    const float* __restrict__ Bp, const float* __restrict__ Bn,
    float* __restrict__ C1, float* __restrict__ Cg1)
{
  int idx = blockIdx.x * 256 + threadIdx.x;   // 0..4095
  int which = idx >> 11;                      // 0 -> C1, 1 -> Cg1
  int j = (idx >> 5) & 63;
  int r = idx & 31;
  const float* wrow = (which ? G1 : W1) + (size_t)j * (2 * DIM) + DIM;
  const float* bm = (r < 16) ? Bp : Bn;
  int rr = r & 15;
  float s = 0.0f;
  for (int d = 0; d < DIM; ++d) s += wrow[d] * bm[d * 16 + rr];
  (which ? Cg1 : C1)[j * 32 + r] = s;
}

// ---------------------------------------------------------------------------
// Kernel B: per token, fused K=4096 contraction producing
//   u = x @ [A_pos;A_neg]^T (32), hbase = base @ W1a^T (64), gbase = base @ G1a^T (64)
// then small MLP tails -> cvec[N][96] = [relu(h) | 2*u], alphas[N]
// 128 threads (4 waves), 64 tokens per WG.
// ---------------------------------------------------------------------------
__global__ __launch_bounds__(128) void phase1_kernel(
    const float* __restrict__ x, const float* __restrict__ base,
    const float* __restrict__ Ap, const float* __restrict__ An,
    const float* __restrict__ W1, const float* __restrict__ b1,
    const float* __restrict__ G1, const float* __restrict__ g1,
    const float* __restrict__ G2, const float* __restrict__ g2,
    const float* __restrict__ C1, const float* __restrict__ Cg1,
    float* __restrict__ cvec, float* __restrict__ alphas)
{
  __shared__ float smem[NCOL * WSTR];   // 160*68*4 = 43,520 B

  const int tid  = threadIdx.x;
  const int lane = tid & 31;
  const int wave = tid >> 5;
  const int t0   = blockIdx.x * 64;
  const int col  = lane & 15;
  const int kk   = (lane >> 4) << 1;           // 0 or 2 (K sub-offset per A/B layout)
  const int trow = t0 + wave * 16 + col;       // this lane's A-matrix row (token)

  const float* xrow = x    + (size_t)trow * DIM;
  const float* brow = base + (size_t)trow * DIM;

  v8f u_acc[2] = {};
  v8f h_acc[4] = {};
  v8f g_acc[4] = {};

  for (int k0 = 0; k0 < DIM; k0 += KC) {
    // Cooperative stage of the 160 weight columns for this K-chunk into LDS.
    for (int i = tid; i < NCOL * (KC / 4); i += 128) {
      int c = i >> 4;              // KC/4 = 16 float4 per column
      int f = i & 15;
      const float* p;
      if      (c < 16) p = Ap + (size_t)c * DIM;
      else if (c < 32) p = An + (size_t)(c - 16) * DIM;
      else if (c < 96) p = W1 + (size_t)(c - 32) * (2 * DIM);
      else             p = G1 + (size_t)(c - 96) * (2 * DIM);
      float4 v = *(const float4*)(p + k0 + (f << 2));
      *(float4*)&smem[c * WSTR + (f << 2)] = v;
    }
    __syncthreads();

    for (int ks = 0; ks < KC; ks += 4) {
      v2f ax = *(const v2f*)(xrow + k0 + ks + kk);
      v2f ab = *(const v2f*)(brow + k0 + ks + kk);
      const float* wl = &smem[ks + kk];
#pragma unroll
      for (int f = 0; f < 2; ++f) {
        v2f b = *(const v2f*)(wl + (f * 16 + col) * WSTR);
        u_acc[f] = WMMA_F32(ax, b, u_acc[f]);
      }
#pragma unroll
      for (int f = 0; f < 4; ++f) {
        v2f b = *(const v2f*)(wl + (32 + f * 16 + col) * WSTR);
        h_acc[f] = WMMA_F32(ab, b, h_acc[f]);
      }
#pragma unroll
      for (int f = 0; f < 4; ++f) {
        v2f b = *(const v2f*)(wl + (96 + f * 16 + col) * WSTR);
        g_acc[f] = WMMA_F32(ab, b, g_acc[f]);
      }
    }
    __syncthreads();
  }

  // Dump accumulators to LDS as ep[token_in_wg][160]  (reuses weight buffer)
  {
    const int m0 = wave * 16 + ((lane >> 4) * 8);   // C/D layout: lanes>=16 hold M=8..15
#pragma unroll
    for (int f = 0; f < 2; ++f)
#pragma unroll
      for (int r = 0; r < 8; ++r)
        smem[(m0 + r) * NCOL + f * 16 + col] = u_acc[f][r];
#pragma unroll
    for (int f = 0; f < 4; ++f)
#pragma unroll
      for (int r = 0; r < 8; ++r)
        smem[(m0 + r) * NCOL + 32 + f * 16 + col] = h_acc[f][r];
#pragma unroll
    for (int f = 0; f < 4; ++f)
#pragma unroll
      for (int r = 0; r < 8; ++r)
        smem[(m0 + r) * NCOL + 96 + f * 16 + col] = g_acc[f][r];
  }
  __syncthreads();

  // Small per-token MLP tails (64 tokens -> first 64 threads).
  if (tid < 64) {
    const int tg = t0 + tid;
    const float* e = &smem[tid * NCOL];
    float us[32];
#pragma unroll
    for (int r = 0; r < 32; ++r) us[r] = 2.0f * e[r];   // SCALING folded in
    float gdot = 0.0f;
    for (int j = 0; j < 64; ++j) {
      float hv = e[32 + j] + b1[j];
      float gv = e[96 + j] + g1[j];
      const float* c1 = C1  + j * 32;
      const float* cg = Cg1 + j * 32;
#pragma unroll
      for (int r = 0; r < 32; ++r) { hv += us[r] * c1[r]; gv += us[r] * cg[r]; }
      hv = fmaxf(hv, 0.0f);
      gv = fmaxf(gv, 0.0f);
      cvec[(size_t)tg * 96 + j] = hv;
      gdot += gv * G2[j];
    }
#pragma unroll
    for (int r = 0; r < 32; ++r) cvec[(size_t)tg * 96 + 64 + r] = us[r];
    float z = 6.0f * (gdot + g2[0]);
    alphas[tg] = 1.0f / (1.0f + __expf(-z));
  }
}

// ---------------------------------------------------------------------------
// Kernel C: out = base + alpha * (cvec @ [W2 | B_pos | B_neg]^T + b2)
// K = 96 (64 from W2, 16+16 from LoRA B banks). 256 threads: 8 waves,
// each wave a 16-token x 64-col tile. grid = (NTOK/16, 4096/512).
// ---------------------------------------------------------------------------
__global__ __launch_bounds__(256) void phase2_kernel(
    const float* __restrict__ base,
    const float* __restrict__ Bp, const float* __restrict__ Bn,
    const float* __restrict__ W2, const float* __restrict__ b2,
    const float* __restrict__ cvec, const float* __restrict__ alphas,
    float* __restrict__ out)
{
  const int tid  = threadIdx.x;
  const int lane = tid & 31;
  const int wave = tid >> 5;
  const int t0   = blockIdx.x * 16;
  const int d0   = blockIdx.y * 512 + wave * 64;
  const int colr = lane & 15;
  const int kk   = (lane >> 4) << 1;
  const float* crow = cvec + (size_t)(t0 + colr) * 96;

  v8f acc[4] = {};

  // segment 1: W2 [4096 x 64], K = 0..63
  for (int k = 0; k < 64; k += 4) {
    v2f a = *(const v2f*)(crow + k + kk);
#pragma unroll
    for (int f = 0; f < 4; ++f) {
      int c = d0 + f * 16 + colr;
      v2f b = *(const v2f*)(W2 + (size_t)c * 64 + k + kk);
      acc[f] = WMMA_F32(a, b, acc[f]);
    }
  }
  // segment 2: B_pos [4096 x 16], K = 64..79
  for (int k = 0; k < 16; k += 4) {
    v2f a = *(const v2f*)(crow + 64 + k + kk);
#pragma unroll
    for (int f = 0; f < 4; ++f) {
      int c = d0 + f * 16 + colr;
      v2f b = *(const v2f*)(Bp + (size_t)c * 16 + k + kk);
      acc[f] = WMMA_F32(a, b, acc[f]);
    }
  }
  // segment 3: B_neg [4096 x 16], K = 80..95
  for (int k = 0; k < 16; k += 4) {
    v2f a = *(const v2f*)(crow + 80 + k + kk);
#pragma unroll
    for (int f = 0; f < 4; ++f) {
      int c = d0 + f * 16 + colr;
      v2f b = *(const v2f*)(Bn + (size_t)c * 16 + k + kk);
      acc[f] = WMMA_F32(a, b, acc[f]);
    }
  }

  // fused epilogue
  const int mbase = (lane >> 4) * 8;
#pragma unroll
  for (int f = 0; f < 4; ++f) {
    const int c = d0 + f * 16 + colr;
    const float bias = b2[c];
#pragma unroll
    for (int r = 0; r < 8; ++r) {
      const int tok = t0 + mbase + r;
      const float al = alphas[tok];
      const size_t idx = (size_t)tok * DIM + c;
      out[idx] = base[idx] + al * (acc[f][r] + bias);
    }
  }
}

// ---------------------------------------------------------------------------
extern "C" void kernel_launch(void* const* d_in, const int* in_sizes, int n_in,
                              void* d_out, int out_size, void* d_ws, size_t ws_size,
                              hipStream_t stream) {
  (void)in_sizes; (void)n_in; (void)out_size; (void)ws_size;
  const float* x    = (const float*)d_in[0];
  const float* base = (const float*)d_in[1];
  const float* Ap   = (const float*)d_in[2];
  const float* Bp   = (const float*)d_in[3];
  const float* An   = (const float*)d_in[4];
  const float* Bn   = (const float*)d_in[5];
  const float* W1   = (const float*)d_in[6];
  const float* b1   = (const float*)d_in[7];
  const float* W2   = (const float*)d_in[8];
  const float* b2   = (const float*)d_in[9];
  const float* G1   = (const float*)d_in[10];
  const float* g1   = (const float*)d_in[11];
  const float* G2   = (const float*)d_in[12];
  const float* g2   = (const float*)d_in[13];
  float* out = (float*)d_out;

  float* ws     = (float*)d_ws;
  float* C1     = ws;                       // 64*32
  float* Cg1    = ws + 2048;                // 64*32
  float* cvec   = ws + 4096;                // NTOK*96
  float* alphas = ws + 4096 + (size_t)NTOK * 96;  // NTOK

  cmat_kernel<<<16, 256, 0, stream>>>(W1, G1, Bp, Bn, C1, Cg1);
  phase1_kernel<<<NTOK / 64, 128, 0, stream>>>(x, base, Ap, An, W1, b1, G1, g1,
                                               G2, g2, C1, Cg1, cvec, alphas);
  phase2_kernel<<<dim3(NTOK / 16, DIM / 512), 256, 0, stream>>>(
      base, Bp, Bn, W2, b2, cvec, alphas, out);
}